// SimpleSNN_22651657519453
// MI455X (gfx1250) — compile-verified
//
#include <hip/hip_runtime.h>

typedef __attribute__((ext_vector_type(2))) float v2f;
typedef __attribute__((ext_vector_type(8))) float v8f;

#define BETA      0.95f
#define THR       1.0f
#define STEPS     10
#define KDIM      784
#define NDIM      10
#define LDSTRIDE  788   // 784 padded: 8B-aligned rows, no LDS bank conflicts
#define WAVES_PER_BLOCK 8

__global__ __launch_bounds__(256) void snn_wmma_f32_kernel(
    const float* __restrict__ x,   // [B, 784]
    const float* __restrict__ W,   // [10, 784]
    float* __restrict__ out)       // [B, 10]
{
    // ---- Stage W into LDS as a 16x788 zero-padded tile (Wlds[n][k] = W[n][k]) ----
    __shared__ float Wlds[16 * LDSTRIDE];
    const int tid = threadIdx.x;
    for (int i = tid; i < 16 * LDSTRIDE; i += 256) {
        const int n = i / LDSTRIDE;
        const int k = i - n * LDSTRIDE;
        float v = 0.0f;
        if (n < NDIM && k < KDIM) v = W[n * KDIM + k];
        Wlds[i] = v;
    }
    __syncthreads();

    // ---- Per-wave 16x16 output tile ----
    const int wave  = tid >> 5;
    const int lane  = tid & 31;
    const int half  = lane >> 4;   // 0: lanes 0-15, 1: lanes 16-31
    const int mn    = lane & 15;   // A: row-in-tile ; B: col (N)
    const int mbase = (blockIdx.x * WAVES_PER_BLOCK + wave) * 16;

    // A (16x4 f32): lane(m, half) VGPR0/1 = x[m][k + 2*half + {0,1}] -> float2 load
    const float* xptr = x + (size_t)(mbase + mn) * KDIM + 2 * half;
    // B (4x16 f32): lane(n, half) VGPR0/1 = W[n][k + 2*half + {0,1}]  (B = W^T, col-major by N)
    const float* bptr = &Wlds[mn * LDSTRIDE + 2 * half];

    v8f c = {0.f, 0.f, 0.f, 0.f, 0.f, 0.f, 0.f, 0.f};

#pragma unroll 4
    for (int k = 0; k < KDIM; k += 4) {
        v2f a = *(const v2f*)(xptr + k);   // global_load_b64
        v2f b = *(const v2f*)(bptr + k);   // ds_load_b64
        // D = A(16x4) * B(4x16) + C ; 8 args: neg_a, A, neg_b, B, c_mod, C, reuse_a, reuse_b
        c = __builtin_amdgcn_wmma_f32_16x16x4_f32(false, a, false, b, (short)0, c,
                                                  false, false);
    }

    // ---- Leaky integrate-and-fire (reset-by-subtraction), 10 steps, spike count ----
    // c[v] element maps to row = mbase + v + 8*half, col = mn
    if (mn < NDIM) {
#pragma unroll
        for (int v = 0; v < 8; ++v) {
            const float cur = c[v];
            float mem = 0.0f, spk = 0.0f, sum = 0.0f;
#pragma unroll
            for (int t = 0; t < STEPS; ++t) {
                mem = BETA * mem + cur - spk * THR;
                spk = (mem > THR) ? 1.0f : 0.0f;
                sum += spk;
            }
            out[(size_t)(mbase + v + 8 * half) * NDIM + mn] = sum;
        }
    }
}

extern "C" void kernel_launch(void* const* d_in, const int* in_sizes, int n_in,
                              void* d_out, int out_size, void* d_ws, size_t ws_size,
                              hipStream_t stream) {
    const float* x = (const float*)d_in[0];   // [B, 784] f32
    const float* W = (const float*)d_in[1];   // [10, 784] f32
    float* out = (float*)d_out;               // [B, 10] f32

    const int B = in_sizes[0] / KDIM;                 // 65536
    const int tilesM = B / 16;                        // 4096
    const int blocks = tilesM / WAVES_PER_BLOCK;      // 512

    snn_wmma_f32_kernel<<<dim3(blocks), dim3(256), 0, stream>>>(x, W, out);
}